// GPAdapter_60181081751932
// MI455X (gfx1250) — compile-verified
//
#include <hip/hip_runtime.h>
#include <hip/hip_bf16.h>

// ---------------------------------------------------------------------------
// GP adapter pipeline for MI455X (gfx1250, wave32, WMMA).
// FP32 everywhere (Cholesky is precision-sensitive); matrix work uses
// V_WMMA_F32_16X16X4_F32.  SYRK K-panel staging uses the CDNA5 async
// global->LDS copy path (ASYNCcnt) when the toolchain exposes it.
// ---------------------------------------------------------------------------

typedef __attribute__((ext_vector_type(2))) float v2f;
typedef __attribute__((ext_vector_type(8))) float v8f;

__device__ __forceinline__ v8f wmma4(v2f a, v2f b, v8f c) {
  // D = A(16x4) * B(4x16) + C(16x16), fp32
  return __builtin_amdgcn_wmma_f32_16x16x4_f32(
      false, a, false, b, (short)0, c, false, false);
}

#if defined(__has_builtin)
#if __has_builtin(__builtin_amdgcn_global_load_async_to_lds_b128) && \
    __has_builtin(__builtin_amdgcn_s_wait_asynccnt)
#define USE_ASYNC_LDS 1
#endif
#endif
#ifndef USE_ASYNC_LDS
#define USE_ASYNC_LDS 0
#endif

#if USE_ASYNC_LDS
// Builtin signature (from clang diagnostic): param0 is a non-const pointer to
// int __attribute__((vector_size(16))) in the global (__device__/AS1) address
// space; param1 is the LDS-side pointer (AS3).
typedef int v4i_async __attribute__((vector_size(16)));
typedef v4i_async __attribute__((address_space(1)))* gas_ptr;
typedef v4i_async __attribute__((address_space(3)))* las_ptr;
#endif

namespace cfg {
constexpr int Bx   = 32;    // batches
constexpr int N    = 1024;  // train points
constexpr int Ts   = 1024;  // test points
constexpr int NMC  = 10;    // monte-carlo samples
constexpr int CH   = 16;    // channels
constexpr int NCLS = 10;    // classes
constexpr int NT   = 17;    // num tasks
constexpr int RK   = 3;     // index-kernel rank
constexpr float JITTER = 1e-4f;
}

// ---------------------------------------------------------------------------
// 0) Btask = W W^T + diag(softplus(v))   [17x17]
// ---------------------------------------------------------------------------
__global__ void btask_kernel(const float* __restrict__ W,
                             const float* __restrict__ v,
                             float* __restrict__ Btask) {
  int tid = threadIdx.x;
  if (tid >= cfg::NT * cfg::NT) return;
  int i = tid / cfg::NT, j = tid % cfg::NT;
  float s = 0.f;
#pragma unroll
  for (int r = 0; r < cfg::RK; ++r) s += W[i * cfg::RK + r] * W[j * cfg::RK + r];
  if (i == j) s += logf(1.f + __expf(v[i]));  // softplus
  Btask[tid] = s;
}

// ---------------------------------------------------------------------------
// 1) Build Kxx (with noise^2+jitter on diag), KsxT (T x Ts, transposed), Kss
//    gridDim = (N*N/256, 3, B)
// ---------------------------------------------------------------------------
__global__ void build_kernel(const float* __restrict__ inputs,
                             const float* __restrict__ test_inputs,
                             const int* __restrict__ indices,
                             const int* __restrict__ test_indices,
                             const float* __restrict__ lengthscale,
                             const float* __restrict__ noise,
                             const float* __restrict__ BtaskG,
                             float* __restrict__ Kxx,
                             float* __restrict__ KsxT,
                             float* __restrict__ Kss) {
  using namespace cfg;
  __shared__ float Bt[NT * NT];
  int tid = threadIdx.x;
  for (int q = tid; q < NT * NT; q += 256) Bt[q] = BtaskG[q];
  __syncthreads();
  int b = blockIdx.z, mat = blockIdx.y;
  size_t e = (size_t)blockIdx.x * 256 + tid;
  int i = (int)(e >> 10), j = (int)(e & 1023);
  float inv_ls = 1.f / lengthscale[0];
  if (mat == 0) {
    float xi = inputs[b * N + i], xj = inputs[b * N + j];
    int ci = indices[b * N + i], cj = indices[b * N + j];
    float d = (xi - xj) * inv_ls;
    float val = __expf(-0.5f * d * d) * Bt[ci * NT + cj];
    if (i == j) { float nz = noise[0]; val += nz * nz + JITTER; }
    Kxx[(size_t)b * N * N + e] = val;
  } else if (mat == 1) {  // KsxT[t=i][s=j] = rbf(xs_j, x_i) * B[i_te_j, i_tr_i]
    float xi = inputs[b * N + i], xj = test_inputs[b * Ts + j];
    int ci = indices[b * N + i], cj = test_indices[b * Ts + j];
    float d = (xi - xj) * inv_ls;
    KsxT[(size_t)b * N * Ts + e] = __expf(-0.5f * d * d) * Bt[cj * NT + ci];
  } else {
    float xi = test_inputs[b * Ts + i], xj = test_inputs[b * Ts + j];
    int ci = test_indices[b * Ts + i], cj = test_indices[b * Ts + j];
    float d = (xi - xj) * inv_ls;
    Kss[(size_t)b * Ts * Ts + e] = __expf(-0.5f * d * d) * Bt[ci * NT + cj];
  }
}

// ---------------------------------------------------------------------------
// 2) Blocked in-place Cholesky (lower).  One workgroup (256 thr = 8 waves)
//    per batch.  16-wide panels; trailing rank-16 update via WMMA f32.
// ---------------------------------------------------------------------------
__global__ void chol_kernel(float* __restrict__ Abase, float diag_add) {
  using namespace cfg;
  constexpr int NBLK = N / 16;
  float* A = Abase + (size_t)blockIdx.x * N * N;
  __shared__ float Dl[16][17];
  int tid = threadIdx.x;
  int lane = tid & 31, wave = tid >> 5;
  int m = lane & 15, kp = lane >> 4;

  for (int kb = 0; kb < NBLK; ++kb) {
    // --- load + factor diagonal block in LDS ---
    {
      int i = tid >> 4, j = tid & 15;
      float val = A[(size_t)(kb * 16 + i) * N + kb * 16 + j];
      if (i == j) val += diag_add;
      Dl[i][j] = val;
    }
    __syncthreads();
    for (int kk = 0; kk < 16; ++kk) {
      if (tid == 0) Dl[kk][kk] = sqrtf(Dl[kk][kk]);
      __syncthreads();
      if (tid > kk && tid < 16) Dl[tid][kk] /= Dl[kk][kk];
      __syncthreads();
      {
        int i = tid >> 4, j = tid & 15;
        if (i > kk && j > kk && j <= i) Dl[i][j] -= Dl[i][kk] * Dl[j][kk];
      }
      __syncthreads();
    }
    {
      int i = tid >> 4, j = tid & 15;
      A[(size_t)(kb * 16 + i) * N + kb * 16 + j] = Dl[i][j];
    }
    __syncthreads();

    // --- panel TRSM: rows below solve against diag block ---
    for (int r = (kb + 1) * 16 + tid; r < N; r += 256) {
      float x[16];
#pragma unroll
      for (int p = 0; p < 16; ++p) x[p] = A[(size_t)r * N + kb * 16 + p];
#pragma unroll
      for (int kk = 0; kk < 16; ++kk) {
        float s = x[kk];
        for (int p = 0; p < kk; ++p) s -= Dl[kk][p] * x[p];
        x[kk] = s / Dl[kk][kk];
      }
#pragma unroll
      for (int p = 0; p < 16; ++p) A[(size_t)r * N + kb * 16 + p] = x[p];
    }
    __syncthreads();

    // --- trailing update: A[ib][jb] -= P_ib * P_jb^T  (WMMA, wave-uniform) ---
    int cnt = 0;
    for (int ib = kb + 1; ib < NBLK; ++ib) {
      for (int jb = kb + 1; jb <= ib; ++jb, ++cnt) {
        if ((cnt & 7) != wave) continue;
        v8f acc = {};
#pragma unroll
        for (int kk = 0; kk < 16; kk += 4) {
          const float* pa = A + (size_t)(ib * 16 + m) * N + kb * 16 + kk + 2 * kp;
          const float* pb = A + (size_t)(jb * 16 + m) * N + kb * 16 + kk + 2 * kp;
          v2f a; a.x = pa[0]; a.y = pa[1];      // A[m][k0], A[m][k0+1]
          v2f bt; bt.x = pb[0]; bt.y = pb[1];   // B[k][n] = P_jb[n][k]
          acc = wmma4(a, bt, acc);
        }
#pragma unroll
        for (int r = 0; r < 8; ++r) {
          int M = r + 8 * kp;
          A[(size_t)(ib * 16 + M) * N + jb * 16 + m] -= acc[r];
        }
      }
    }
    __syncthreads();
  }
}

// ---------------------------------------------------------------------------
// 3) Blocked TRSM:  X <- L^{-1} X   (X = KsxT, 1024 x 1024, in place)
//    grid (Ts/128, B); 8 waves, each owns 16 columns.  WMMA GEMM update,
//    LDS forward substitution.
// ---------------------------------------------------------------------------
__global__ void trsm_kernel(const float* __restrict__ Lbase,
                            float* __restrict__ Xbase) {
  using namespace cfg;
  constexpr int NBLK = N / 16;
  int b = blockIdx.y;
  const float* L = Lbase + (size_t)b * N * N;
  float* X = Xbase + (size_t)b * N * Ts;
  __shared__ float Dl[16][17];
  __shared__ float U[16][130];
  int tid = threadIdx.x, lane = tid & 31, wave = tid >> 5;
  int m = lane & 15, kp = lane >> 4;
  int col0 = blockIdx.x * 128 + wave * 16;

  for (int kb = 0; kb < NBLK; ++kb) {
    {
      int i = tid >> 4, j = tid & 15;
      Dl[i][j] = L[(size_t)(kb * 16 + i) * N + kb * 16 + j];
    }
    v8f acc = {};
    for (int p = 0; p < kb; ++p) {
#pragma unroll
      for (int kk = 0; kk < 16; kk += 4) {
        const float* pa = L + (size_t)(kb * 16 + m) * N + p * 16 + kk + 2 * kp;
        v2f a; a.x = pa[0]; a.y = pa[1];
        int k0 = p * 16 + kk + 2 * kp;
        v2f bv;
        bv.x = X[(size_t)k0 * Ts + col0 + m];
        bv.y = X[(size_t)(k0 + 1) * Ts + col0 + m];
        acc = wmma4(a, bv, acc);
      }
    }
#pragma unroll
    for (int r = 0; r < 8; ++r) U[r + 8 * kp][wave * 16 + m] = acc[r];
    __syncthreads();

    if (tid < 128) {
      int c = tid, col = blockIdx.x * 128 + c;
#pragma unroll
      for (int kk = 0; kk < 16; ++kk) {
        float s = X[(size_t)(kb * 16 + kk) * Ts + col] - U[kk][c];
        for (int p = 0; p < kk; ++p) s -= Dl[kk][p] * U[p][c];
        U[kk][c] = s / Dl[kk][kk];
      }
#pragma unroll
      for (int kk = 0; kk < 16; ++kk)
        X[(size_t)(kb * 16 + kk) * Ts + col] = U[kk][c];
    }
    __syncthreads();
  }
}

// ---------------------------------------------------------------------------
// 4) SYRK:  cov = Kss - V^T V   (in place on Kss).  Lower-tri 64x64 macro
//    tiles; grid (136, B).  K panels (32 deep) staged into LDS via the
//    CDNA5 async global->LDS path (ASYNCcnt) when available; 2 WMMA tiles
//    per wave.  Row stride 72 floats: 16B-aligned rows for B128-to-LDS and
//    conflict-free fragment reads (2*72 mod 64 == 16).
// ---------------------------------------------------------------------------
__global__ void syrk_kernel(const float* __restrict__ Vbase,
                            float* __restrict__ Cbase) {
  using namespace cfg;
  constexpr int LSTR = 72;
  int b = blockIdx.y;
  const float* V = Vbase + (size_t)b * N * Ts;
  float* Cg = Cbase + (size_t)b * Ts * Ts;
  __shared__ float As[32][LSTR];
  __shared__ float Bs[32][LSTR];

  int t = blockIdx.x, I = 0;
  while (t >= I + 1) { t -= I + 1; ++I; }
  int J = t;  // J <= I

  int tid = threadIdx.x, lane = tid & 31, wave = tid >> 5;
  int m = lane & 15, kp = lane >> 4;
  int t0 = wave, t1 = wave + 8;
  int tr0 = t0 >> 2, tc0 = t0 & 3, tr1 = t1 >> 2, tc1 = t1 & 3;
  v8f acc0 = {}, acc1 = {};

  for (int kb = 0; kb < N; kb += 32) {
    __syncthreads();
#if USE_ASYNC_LDS
    // 512 x 16B chunks per panel; 2 chunks/thread/panel, no VGPR staging.
#pragma unroll
    for (int q = 0; q < 2; ++q) {
      int e = tid + q * 256;          // chunk id
      int r = e >> 4;                 // row (0..31)
      int c4 = (e & 15) << 2;         // col (multiple of 4 floats)
      __builtin_amdgcn_global_load_async_to_lds_b128(
          (gas_ptr)(V + (size_t)(kb + r) * Ts + I * 64 + c4),
          (las_ptr)&As[r][c4], 0, 0);
      __builtin_amdgcn_global_load_async_to_lds_b128(
          (gas_ptr)(V + (size_t)(kb + r) * Ts + J * 64 + c4),
          (las_ptr)&Bs[r][c4], 0, 0);
    }
    __builtin_amdgcn_s_wait_asynccnt(0);
#else
#pragma unroll
    for (int q = 0; q < 8; ++q) {
      int e = tid + q * 256;
      int r = e >> 6, c2 = e & 63;
      As[r][c2] = V[(size_t)(kb + r) * Ts + I * 64 + c2];
      Bs[r][c2] = V[(size_t)(kb + r) * Ts + J * 64 + c2];
    }
#endif
    if (kb + 32 < N)  // hint next K panel toward the caches
      __builtin_prefetch(&V[(size_t)(kb + 32 + (tid >> 3)) * Ts + I * 64], 0, 1);
    __syncthreads();
#pragma unroll
    for (int kk = 0; kk < 32; kk += 4) {
      int k0 = kk + 2 * kp;
      v2f a0, b0, a1, b1;
      a0.x = As[k0][tr0 * 16 + m]; a0.y = As[k0 + 1][tr0 * 16 + m];
      b0.x = Bs[k0][tc0 * 16 + m]; b0.y = Bs[k0 + 1][tc0 * 16 + m];
      acc0 = wmma4(a0, b0, acc0);
      a1.x = As[k0][tr1 * 16 + m]; a1.y = As[k0 + 1][tr1 * 16 + m];
      b1.x = Bs[k0][tc1 * 16 + m]; b1.y = Bs[k0 + 1][tc1 * 16 + m];
      acc1 = wmma4(a1, b1, acc1);
    }
  }
#pragma unroll
  for (int r = 0; r < 8; ++r) {
    int M = r + 8 * kp;
    {
      size_t o = (size_t)(I * 64 + tr0 * 16 + M) * Ts + J * 64 + tc0 * 16 + m;
      Cg[o] = Cg[o] - acc0[r];
    }
    {
      size_t o = (size_t)(I * 64 + tr1 * 16 + M) * Ts + J * 64 + tc1 * 16 + m;
      Cg[o] = Cg[o] - acc1[r];
    }
  }
}

// ---------------------------------------------------------------------------
// 5) w = L^{-1}(y - c)   forward substitution, column sweep.  grid (B).
// ---------------------------------------------------------------------------
__global__ void fsolve_kernel(const float* __restrict__ Lbase,
                              const float* __restrict__ values,
                              const float* __restrict__ mean_const,
                              float* __restrict__ wbuf) {
  using namespace cfg;
  int b = blockIdx.x;
  const float* L = Lbase + (size_t)b * N * N;
  __shared__ float r[N];
  int tid = threadIdx.x;
  float c0 = mean_const[0];
  for (int q = tid; q < N; q += 256) r[q] = values[b * N + q] - c0;
  __syncthreads();
  for (int tcol = 0; tcol < N; ++tcol) {
    if (tid == 0) r[tcol] /= L[(size_t)tcol * N + tcol];
    __syncthreads();
    float wt = r[tcol];
    for (int i = tcol + 1 + tid; i < N; i += 256)
      r[i] -= L[(size_t)i * N + tcol] * wt;
    __syncthreads();
  }
  for (int q = tid; q < N; q += 256) wbuf[b * N + q] = r[q];
}

// ---------------------------------------------------------------------------
// 6) mean[b][s] = c + sum_t V[t][s] * w[t]     grid (Ts/256, B)
// ---------------------------------------------------------------------------
__global__ void mean_kernel(const float* __restrict__ Vbase,
                            const float* __restrict__ wbuf,
                            const float* __restrict__ mean_const,
                            float* __restrict__ meanb) {
  using namespace cfg;
  int b = blockIdx.y;
  const float* V = Vbase + (size_t)b * N * Ts;
  __shared__ float wch[256];
  int tid = threadIdx.x;
  int s = blockIdx.x * 256 + tid;
  float acc = mean_const[0];
  for (int t0 = 0; t0 < N; t0 += 256) {
    __syncthreads();
    wch[tid] = wbuf[b * N + t0 + tid];
    __syncthreads();
    for (int tt = 0; tt < 256; ++tt)
      acc += V[(size_t)(t0 + tt) * Ts + s] * wch[tt];
  }
  meanb[b * Ts + s] = acc;
}

// ---------------------------------------------------------------------------
// 7) samples[m][b][s] = mean[b][s] + sum_{j<=s} Lstar[s][j] * eps[m][b][j]
//    grid (B, NMC), block 1024.
// ---------------------------------------------------------------------------
__global__ void sample_kernel(const float* __restrict__ Lsbase,
                              const float* __restrict__ meanb,
                              const float* __restrict__ eps,
                              float* __restrict__ samp) {
  using namespace cfg;
  int b = blockIdx.x, mcs = blockIdx.y;
  const float* Ls = Lsbase + (size_t)b * Ts * Ts;
  __shared__ float el[Ts];
  int tid = threadIdx.x;
  el[tid] = eps[((size_t)mcs * Bx + b) * Ts + tid];
  __syncthreads();
  float acc = meanb[b * Ts + tid];
  const float* row = Ls + (size_t)tid * Ts;
  for (int j = 0; j <= tid; ++j) acc += row[j] * el[j];
  samp[((size_t)mcs * Bx + b) * Ts + tid] = acc;
}

// ---------------------------------------------------------------------------
// 8) channel scatter + mean pool (/max_len=64) + linear classifier
//    grid (NMC*B), block 256.
// ---------------------------------------------------------------------------
__global__ void feat_kernel(const float* __restrict__ samp,
                            const int* __restrict__ test_idx,
                            const float* __restrict__ W_clf,
                            const float* __restrict__ b_clf,
                            float* __restrict__ out) {
  using namespace cfg;
  int blk = blockIdx.x;
  int mcs = blk / Bx, b = blk % Bx;
  __shared__ float fl[CH];
  int tid = threadIdx.x;
  if (tid < CH) fl[tid] = 0.f;
  __syncthreads();
  for (int s = tid; s < Ts; s += 256) {
    int ch = test_idx[b * Ts + s];
    atomicAdd(&fl[ch], samp[((size_t)mcs * Bx + b) * Ts + s]);
  }
  __syncthreads();
  if (tid < NCLS) {
    float o = b_clf[tid];
    const float inv_len = 1.f / 64.f;  // max_len = Ts / CH
#pragma unroll
    for (int ch = 0; ch < CH; ++ch)
      o += fl[ch] * inv_len * W_clf[ch * NCLS + tid];
    out[((size_t)mcs * Bx + b) * NCLS + tid] = o;
  }
}

// ---------------------------------------------------------------------------
// launcher
// ---------------------------------------------------------------------------
extern "C" void kernel_launch(void* const* d_in, const int* in_sizes, int n_in,
                              void* d_out, int out_size, void* d_ws, size_t ws_size,
                              hipStream_t stream) {
  using namespace cfg;
  (void)in_sizes; (void)n_in; (void)out_size; (void)ws_size;
  const float* inputs       = (const float*)d_in[0];   // [B,T]
  const float* values       = (const float*)d_in[1];   // [B,T]
  const float* test_inputs  = (const float*)d_in[2];   // [B,Ts]
  const float* eps          = (const float*)d_in[3];   // [NMC,B,Ts]
  const float* mean_const   = (const float*)d_in[4];
  const float* lengthscale  = (const float*)d_in[5];
  const float* noise        = (const float*)d_in[6];
  const float* W_task       = (const float*)d_in[7];   // [17,3]
  const float* v_task       = (const float*)d_in[8];   // [17]
  const float* W_clf        = (const float*)d_in[9];   // [16,10]
  const float* b_clf        = (const float*)d_in[10];  // [10]
  const int*   indices      = (const int*)d_in[11];    // [B,T]
  const int*   test_indices = (const int*)d_in[12];    // [B,Ts]
  float* out = (float*)d_out;

  const size_t MM = (size_t)N * N;
  float* wsf   = (float*)d_ws;
  float* Btask = wsf;                          // 289 (pad 512)
  float* Kxx   = wsf + 512;                    // B*MM  -> L
  float* Vm    = Kxx + (size_t)Bx * MM;        // KsxT  -> V
  float* Kss   = Vm + (size_t)Bx * MM;         // Kss -> cov -> Lstar
  float* wbuf  = Kss + (size_t)Bx * MM;        // B*N
  float* meanb = wbuf + (size_t)Bx * N;        // B*Ts
  float* samp  = meanb + (size_t)Bx * Ts;      // NMC*B*Ts

  btask_kernel<<<1, 320, 0, stream>>>(W_task, v_task, Btask);
  build_kernel<<<dim3(N * N / 256, 3, Bx), 256, 0, stream>>>(
      inputs, test_inputs, indices, test_indices, lengthscale, noise, Btask,
      Kxx, Vm, Kss);
  chol_kernel<<<Bx, 256, 0, stream>>>(Kxx, 0.f);                 // Kxx -> L
  trsm_kernel<<<dim3(Ts / 128, Bx), 256, 0, stream>>>(Kxx, Vm);  // KsxT -> V
  syrk_kernel<<<dim3((Ts / 64) * (Ts / 64 + 1) / 2, Bx), 256, 0, stream>>>(
      Vm, Kss);                                                  // Kss -> cov
  chol_kernel<<<Bx, 256, 0, stream>>>(Kss, JITTER);              // cov -> Lstar
  fsolve_kernel<<<Bx, 256, 0, stream>>>(Kxx, values, mean_const, wbuf);
  mean_kernel<<<dim3(Ts / 256, Bx), 256, 0, stream>>>(Vm, wbuf, mean_const, meanb);
  sample_kernel<<<dim3(Bx, NMC), 1024, 0, stream>>>(Kss, meanb, eps, samp);
  feat_kernel<<<NMC * Bx, 256, 0, stream>>>(samp, test_indices, W_clf, b_clf, out);
}